// KNN_25812753449617
// MI455X (gfx1250) — compile-verified
//
#include <hip/hip_runtime.h>
#include <cmath>

#ifndef M_PI
#define M_PI 3.14159265358979323846
#endif

namespace {
constexpr int   SEARCH      = 7;
constexpr int   K2          = SEARCH * SEARCH;   // 49
constexpr int   PAD         = (SEARCH - 1) / 2;  // 3
constexpr int   CENTER      = (K2 - 1) / 2;      // 24
constexpr int   KNN         = 7;
constexpr float CUTOFF      = 2.0f;
constexpr int   NUM_CLASSES = 20;
constexpr int   H           = 128;
constexpr int   W           = 2048;

struct InvG { float w[K2]; };
} // namespace

__global__ __launch_bounds__(256) void knn_post_kernel(
    const float* __restrict__ proj_range,    // [H*W]
    const float* __restrict__ unproj_range,  // [P]
    const int*   __restrict__ proj_argmax,   // [H*W]
    const int*   __restrict__ px,            // [P]
    const int*   __restrict__ py,            // [P]
    float*       __restrict__ out,           // [P]
    int P, InvG g)
{
  const int p = blockIdx.x * blockDim.x + threadIdx.x;
  if (p >= P) return;

  // gfx1250 global_prefetch_b8: pull the upcoming streamed point data toward us.
  {
    int pn = p + 16384; if (pn >= P) pn = p;
    __builtin_prefetch(px + pn, 0, 0);
    __builtin_prefetch(py + pn, 0, 0);
    __builtin_prefetch(unproj_range + pn, 0, 0);
  }

  const int   cx = px[p];
  const int   cy = py[p];
  const float ur = unproj_range[p];

  // 7 smallest keys, sorted ascending. key = (f32bits(d) << 32) | k.
  // d >= 0 always => u64 order == lexicographic (d, k) == top_k tie-break.
  unsigned long long best[KNN];
#pragma unroll
  for (int j = 0; j < KNN; ++j) best[j] = ~0ull;

#pragma unroll
  for (int k = 0; k < K2; ++k) {
    const int r = cy + (k / SEARCH) - PAD;
    const int c = cx + (k % SEARCH) - PAD;
    const bool inb = ((unsigned)r < (unsigned)H) && ((unsigned)c < (unsigned)W);
    const int rcl = (r < 0) ? 0 : ((r > H - 1) ? H - 1 : r);
    const int ccl = (c < 0) ? 0 : ((c > W - 1) ? W - 1 : c);
    float v = proj_range[rcl * W + ccl];      // always-in-bounds clamped load
    v = inb ? v : 0.0f;                       // jnp.pad zero padding
    v = (v < 0.0f) ? __uint_as_float(0x7f800000u) : v;  // invalid -> +inf
    const float d = (k == CENTER) ? 0.0f : (fabsf(v - ur) * g.w[k]);

    unsigned long long key =
        ((unsigned long long)__float_as_uint(d) << 32) | (unsigned)k;
    // sorted insertion; strict u64 '<' keeps earlier indices on ties
#pragma unroll
    for (int j = 0; j < KNN; ++j) {
      const unsigned long long a = best[j];
      const bool lt = key < a;
      best[j] = lt ? key : a;
      key     = lt ? a : key;
    }
  }

  // Vote among classes 1..19 using 3-bit packed counters (max count 7 fits).
  unsigned long long votes = 0ull;
#pragma unroll
  for (int j = 0; j < KNN; ++j) {
    const unsigned kk   = (unsigned)(best[j] & 0xffffffffull);
    const float    dist = __uint_as_float((unsigned)(best[j] >> 32));
    const int r = cy + (int)(kk / (unsigned)SEARCH) - PAD;
    const int c = cx + (int)(kk % (unsigned)SEARCH) - PAD;
    const bool inb = ((unsigned)r < (unsigned)H) && ((unsigned)c < (unsigned)W);
    const int rcl = (r < 0) ? 0 : ((r > H - 1) ? H - 1 : r);
    const int ccl = (c < 0) ? 0 : ((c > W - 1) ? W - 1 : c);
    int lab = proj_argmax[rcl * W + ccl];
    lab = inb ? lab : 0;                         // zero-padded argmax
    lab = (dist > CUTOFF) ? NUM_CLASSES : lab;   // cutoff -> class 20
    const bool valid = (lab >= 1) && (lab < NUM_CLASSES);
    const unsigned sh = ((unsigned)(3 * (lab - 1))) & 63u;  // defined shift
    votes += valid ? (1ull << sh) : 0ull;
  }

  // argmax over classes 1..19; strict '>' ascending == jnp.argmax first-max.
  int bestC   = 1;
  int bestCnt = (int)(votes & 7ull);
#pragma unroll
  for (int c = 2; c < NUM_CLASSES; ++c) {
    const int cnt = (int)((votes >> (3 * (c - 1))) & 7ull);
    const bool gt = cnt > bestCnt;
    bestCnt = gt ? cnt : bestCnt;
    bestC   = gt ? c : bestC;
  }

  __builtin_nontemporal_store((float)bestC, &out[p]);
}

extern "C" void kernel_launch(void* const* d_in, const int* in_sizes, int n_in,
                              void* d_out, int out_size, void* d_ws, size_t ws_size,
                              hipStream_t stream) {
  (void)n_in; (void)out_size; (void)d_ws; (void)ws_size;

  const float* proj_range   = (const float*)d_in[0];
  const float* unproj_range = (const float*)d_in[1];
  const int*   proj_argmax  = (const int*)d_in[2];
  const int*   px           = (const int*)d_in[3];
  const int*   py           = (const int*)d_in[4];
  float*       out          = (float*)d_out;
  const int    P            = in_sizes[1];

  // 1 - normalized 2D Gaussian (sigma=1), computed in double like the reference.
  InvG g;
  double gg[K2];
  double s = 0.0;
  for (int y = 0; y < SEARCH; ++y) {
    for (int x = 0; x < SEARCH; ++x) {
      const double dx = (double)x - (SEARCH - 1) / 2.0;
      const double dy = (double)y - (SEARCH - 1) / 2.0;
      const double val = std::exp(-(dx * dx + dy * dy) * 0.5) / (2.0 * M_PI);
      gg[y * SEARCH + x] = val;
      s += val;
    }
  }
  for (int i = 0; i < K2; ++i) g.w[i] = (float)(1.0 - gg[i] / s);

  const dim3 block(256);
  const dim3 grid((unsigned)((P + 255) / 256));
  knn_post_kernel<<<grid, block, 0, stream>>>(
      proj_range, unproj_range, proj_argmax, px, py, out, P, g);
}